// GCN_38981123179101
// MI455X (gfx1250) — compile-verified
//
#include <hip/hip_runtime.h>
#include <hip/hip_bf16.h>
#include <math.h>

typedef __attribute__((ext_vector_type(2))) float v2f;
typedef __attribute__((ext_vector_type(8))) float v8f;

#define D_IN  128
#define D_HID 16
#define N_CLS 2
#define GEMM_WAVES 4

// ---------------- degree / normalization ----------------
__global__ void k_deg_init(float* deg, int n) {
    int i = blockIdx.x * blockDim.x + threadIdx.x;
    if (i < n) deg[i] = 1.0f;  // self-loop contributes 1 to every node's degree
}

__global__ void k_deg_edges(const long long* __restrict__ dst, float* deg, int e) {
    int i = blockIdx.x * blockDim.x + threadIdx.x;
    if (i < e) atomicAdd(&deg[(int)dst[i]], 1.0f);
}

__global__ void k_dinv(float* deg, int n) {
    int i = blockIdx.x * blockDim.x + threadIdx.x;
    if (i < n) deg[i] = rsqrtf(deg[i]);  // deg >= 1 always (self-loops)
}

// ---------------- layer-1 GEMM: h1 = x @ W1 via V_WMMA_F32_16X16X4_F32 ----------------
// One wave per 16-row tile, N=16 = D_HID, K looped over 128 in steps of 4.
__global__ __launch_bounds__(GEMM_WAVES * 32)
void k_gemm1_wmma(const float* __restrict__ x, const float* __restrict__ W1,
                  float* __restrict__ h1, int nTiles) {
    __shared__ float wt[D_HID * D_IN];            // W1 transposed: [n][k]
    __shared__ float xs[GEMM_WAVES][16 * D_IN];   // per-wave A tile: [m][k]

    const int tid = threadIdx.x;
    // cooperative load of W1^T (2048 floats)
    for (int idx = tid; idx < D_HID * D_IN; idx += GEMM_WAVES * 32) {
        int nn = idx >> 7;          // [0,16)
        int kk = idx & (D_IN - 1);  // [0,128)
        wt[idx] = W1[kk * D_HID + nn];
    }

    const int wave = tid >> 5;
    const int lane = tid & 31;
    const int tile = blockIdx.x * GEMM_WAVES + wave;

    if (tile < nTiles) {
        // coalesced float4 load of the 16x128 A tile (rows are contiguous)
        const float* gx = x + (size_t)tile * 16 * D_IN;
        #pragma unroll
        for (int i = 0; i < 16; ++i) {
            int off = i * D_IN + lane * 4;
            *(float4*)&xs[wave][off] = *(const float4*)(gx + off);
        }
    }
    __syncthreads();
    if (tile >= nTiles) return;

    // wave32 f32 WMMA lane layout: lanes 0-15 hold K+0,K+1; lanes 16-31 hold K+2,K+3
    const int half = lane >> 4;      // 0 or 1
    const int mn   = lane & 15;      // A row / B col
    const float* xrow = &xs[wave][mn * D_IN];
    const float* wrow = &wt[mn * D_IN];

    v8f acc = {0.f, 0.f, 0.f, 0.f, 0.f, 0.f, 0.f, 0.f};
    #pragma unroll
    for (int k = 0; k < D_IN; k += 4) {
        const int kb = k + half * 2;
        v2f a = *(const v2f*)(xrow + kb);   // A[m][kb], A[m][kb+1]
        v2f b = *(const v2f*)(wrow + kb);   // B[kb][n], B[kb+1][n]
        acc = __builtin_amdgcn_wmma_f32_16x16x4_f32(
            /*neg_a=*/false, a, /*neg_b=*/false, b,
            /*c_mod=*/(short)0, acc, /*reuse_a=*/false, /*reuse_b=*/false);
    }

    // D layout: lanes 0-15 -> rows v, lanes 16-31 -> rows v+8; col = mn
    const int rowBase = tile * 16 + half * 8;
    #pragma unroll
    for (int v = 0; v < 8; ++v)
        h1[(size_t)(rowBase + v) * D_HID + mn] = acc[v];
}

// ---------------- self-loop init: out[i][:] = dinv[i]^2 * h[i][:] ----------------
__global__ void k_selfloop(const float* __restrict__ h, const float* __restrict__ dinv,
                           float* __restrict__ out, int total, int d) {
    int idx = blockIdx.x * blockDim.x + threadIdx.x;
    if (idx >= total) return;
    int i = idx / d;
    float di = dinv[i];
    out[idx] = di * di * h[idx];
}

// ---------------- edge scatter-add: out[dst] += dinv[s]*dinv[d] * h[src] ----------------
__global__ void k_scatter(const long long* __restrict__ src, const long long* __restrict__ dst,
                          const float* __restrict__ dinv, const float* __restrict__ h,
                          float* out, int e, int d) {
    int i = blockIdx.x * blockDim.x + threadIdx.x;
    if (i >= e) return;
    int s = (int)src[i];
    int t = (int)dst[i];
    float nrm = dinv[s] * dinv[t];
    const float* hs = h + (size_t)s * d;
    float* ot = out + (size_t)t * d;
    for (int f = 0; f < d; ++f)
        atomicAdd(ot + f, nrm * hs[f]);
}

// ---------------- layer 2: h2 = relu(out1 + b1) @ W2  (16 -> 2) ----------------
__global__ void k_layer2(const float* __restrict__ out1, const float* __restrict__ b1,
                         const float* __restrict__ W2, float* __restrict__ h2, int n) {
    int i = blockIdx.x * blockDim.x + threadIdx.x;
    if (i >= n) return;
    float z0 = 0.f, z1 = 0.f;
    const float* row = out1 + (size_t)i * D_HID;
    #pragma unroll
    for (int f = 0; f < D_HID; ++f) {
        float v = fmaxf(row[f] + b1[f], 0.f);
        z0 = fmaf(v, W2[f * N_CLS + 0], z0);
        z1 = fmaf(v, W2[f * N_CLS + 1], z1);
    }
    h2[(size_t)i * N_CLS + 0] = z0;
    h2[(size_t)i * N_CLS + 1] = z1;
}

// ---------------- final bias + log_softmax over 2 classes ----------------
__global__ void k_logsoftmax(const float* __restrict__ out2, const float* __restrict__ b2,
                             float* __restrict__ y, int n) {
    int i = blockIdx.x * blockDim.x + threadIdx.x;
    if (i >= n) return;
    float z0 = out2[(size_t)i * N_CLS + 0] + b2[0];
    float z1 = out2[(size_t)i * N_CLS + 1] + b2[1];
    float m = fmaxf(z0, z1);
    float lse = m + logf(expf(z0 - m) + expf(z1 - m));
    y[(size_t)i * N_CLS + 0] = z0 - lse;
    y[(size_t)i * N_CLS + 1] = z1 - lse;
}

extern "C" void kernel_launch(void* const* d_in, const int* in_sizes, int n_in,
                              void* d_out, int out_size, void* d_ws, size_t ws_size,
                              hipStream_t stream) {
    (void)n_in; (void)out_size; (void)ws_size;
    const float*     x   = (const float*)d_in[0];
    const long long* ei  = (const long long*)d_in[1];   // int64 [2, E]
    const float*     W1  = (const float*)d_in[2];
    const float*     b1  = (const float*)d_in[3];
    const float*     W2  = (const float*)d_in[4];
    const float*     b2  = (const float*)d_in[5];

    const int n = in_sizes[0] / D_IN;   // 100000
    const int e = in_sizes[1] / 2;      // 3200000
    const long long* src = ei;
    const long long* dst = ei + e;

    // workspace partition (floats)
    float* ws   = (float*)d_ws;
    float* dinv = ws;  ws += n;
    float* h1   = ws;  ws += (size_t)n * D_HID;
    float* out1 = ws;  ws += (size_t)n * D_HID;
    float* h2   = ws;  ws += (size_t)n * N_CLS;
    float* out2 = ws;  ws += (size_t)n * N_CLS;

    const int B = 256;
    const int gN  = (n + B - 1) / B;
    const int gE  = (e + B - 1) / B;

    // normalization coefficients
    k_deg_init <<<gN, B, 0, stream>>>(dinv, n);
    k_deg_edges<<<gE, B, 0, stream>>>(dst, dinv, e);
    k_dinv     <<<gN, B, 0, stream>>>(dinv, n);

    // layer 1 GEMM (WMMA): n is a multiple of 16 here (100000 = 6250*16)
    const int nTiles = (n + 15) / 16;
    k_gemm1_wmma<<<(nTiles + GEMM_WAVES - 1) / GEMM_WAVES, GEMM_WAVES * 32, 0, stream>>>(
        x, W1, h1, nTiles);

    // aggregation 1
    k_selfloop<<<((n * D_HID) + B - 1) / B, B, 0, stream>>>(h1, dinv, out1, n * D_HID, D_HID);
    k_scatter <<<gE, B, 0, stream>>>(src, dst, dinv, h1, out1, e, D_HID);

    // layer 2 (+bias +relu folded in)
    k_layer2<<<gN, B, 0, stream>>>(out1, b1, W2, h2, n);

    // aggregation 2
    k_selfloop<<<((n * N_CLS) + B - 1) / B, B, 0, stream>>>(h2, dinv, out2, n * N_CLS, N_CLS);
    k_scatter <<<gE, B, 0, stream>>>(src, dst, dinv, h2, out2, e, N_CLS);

    // bias + log_softmax
    k_logsoftmax<<<gN, B, 0, stream>>>(out2, b2, (float*)d_out, n);
}